// LogFlashAttention1d_35459249996587
// MI455X (gfx1250) — compile-verified
//
#include <hip/hip_runtime.h>
#include <math.h>

typedef __attribute__((ext_vector_type(2))) float v2f;
typedef __attribute__((ext_vector_type(8))) float v8f;

#define BB   4
#define W0L  4096
#define CC   512
#define HH   8
#define DD   64
#define BN_EPS 1e-5f

// base 2: n=12, Wp=4096 ; base 3: n=8, Wp=6561 (row-padded to 6656 = 52*128)
#define LP2  4096
#define WP2  4096
#define N2   12
#define LP3  6656
#define WP3  6561
#define N3   8

#define KC   64   // k-chunk staged in LDS

// ---------------------------------------------------------------------------
// gfx1250 async global->LDS DMA (bypasses VGPRs, tracked by ASYNCcnt)
// ---------------------------------------------------------------------------
__device__ __forceinline__ void async_ld_b128(unsigned lds_off, const float* g) {
  asm volatile("global_load_async_to_lds_b128 %0, %1, off"
               :: "v"(lds_off), "v"((unsigned long long)g)
               : "memory");
}
__device__ __forceinline__ void wait_async0() {
#if __has_builtin(__builtin_amdgcn_s_wait_asynccnt)
  __builtin_amdgcn_s_wait_asynccnt(0);
#else
  asm volatile("s_wait_asynccnt 0x0" ::: "memory");
#endif
}

// ---------------------------------------------------------------------------
// elementwise helpers
// ---------------------------------------------------------------------------
__global__ void copy_f4(float4* __restrict__ dst, const float4* __restrict__ src, int n4) {
  int i = blockIdx.x * blockDim.x + threadIdx.x;
  if (i < n4) dst[i] = src[i];
}

// xp: (BB, LP3, CC) <- xt: (BB, W0L, CC), zero rows l >= W0L
__global__ void pad_rows(float4* __restrict__ xp, const float4* __restrict__ xt) {
  size_t i = (size_t)blockIdx.x * blockDim.x + threadIdx.x;
  const size_t n4 = (size_t)BB * LP3 * (CC / 4);
  if (i >= n4) return;
  int c4 = (int)(i % (CC / 4));
  size_t tmp = i / (CC / 4);
  int l = (int)(tmp % LP3);
  int b = (int)(tmp / LP3);
  float4 z; z.x = z.y = z.z = z.w = 0.f;
  xp[i] = (l < W0L) ? xt[((size_t)b * W0L + l) * (CC / 4) + c4] : z;
}

// ---------------------------------------------------------------------------
// GEMM + BatchNorm:  Y[b,l,o] = BN_o( sum_c X[b,l,c] * Wt[o,c] )
//   X: (BB, Lp, CC) row-major; Wt: (O, CC) row-major; bn: (4, O) = g,b,m,v
//   Y (and optional Y2): (BB, Lp, O)
// Block: 256 threads (8 waves); tile 128(l) x 64(o); WMMA f32 16x16x4.
// Tiles staged via global_load_async_to_lds_b128.
// ---------------------------------------------------------------------------
__global__ void __launch_bounds__(256) gemm_bn(
    const float* __restrict__ X, const float* __restrict__ Wt,
    const float* __restrict__ bn, float* __restrict__ Y, float* __restrict__ Y2,
    int Lp, int O)
{
  __shared__ float Xs[128][KC + 4];
  __shared__ float Ws[64][KC + 4];

  const int b  = blockIdx.z;
  const int l0 = blockIdx.x * 128;
  const int o0 = blockIdx.y * 64;
  const float* Xb = X + ((size_t)b * Lp + l0) * CC;

  const int t  = threadIdx.x;
  const int wv = t >> 5;        // wave id 0..7  -> l sub-tile
  const int ln = t & 31;        // lane id

  v8f acc0 = {0.f,0.f,0.f,0.f,0.f,0.f,0.f,0.f};
  v8f acc1 = acc0, acc2 = acc0, acc3 = acc0;

  const int cid = t & 15;       // float4 column within row
  const int r0  = t >> 4;       // row within 16-row pass
  const int arow = wv * 16 + (ln & 15);
  const int koff = (ln < 16) ? 0 : 2;

  for (int c0 = 0; c0 < CC; c0 += KC) {
    // stage X tile: 128 x 64 floats (8 async b128 per thread, no VGPR round-trip)
#pragma unroll
    for (int p = 0; p < 8; ++p) {
      int row = p * 16 + r0;
      async_ld_b128((unsigned)(size_t)&Xs[row][cid * 4],
                    Xb + (size_t)row * CC + c0 + cid * 4);
    }
    // stage W tile: 64 x 64 floats (4 async b128 per thread)
#pragma unroll
    for (int p = 0; p < 4; ++p) {
      int row = p * 16 + r0;
      async_ld_b128((unsigned)(size_t)&Ws[row][cid * 4],
                    Wt + (size_t)(o0 + row) * CC + c0 + cid * 4);
    }
    if (c0 + KC < CC)  // hint next chunk -> global_prefetch_b8
      __builtin_prefetch(Xb + (size_t)(t >> 1) * CC + c0 + KC + (t & 1) * 32, 0, 0);

    wait_async0();       // my wave's DMA into LDS complete
    __syncthreads();     // everyone's complete

#pragma unroll
    for (int kk = 0; kk < KC; kk += 4) {
      v2f a;
      a.x = Xs[arow][kk + koff];
      a.y = Xs[arow][kk + koff + 1];

      v2f b0, b1, b2, b3;
      int oc = ln & 15;
      b0.x = Ws[oc +  0][kk + koff]; b0.y = Ws[oc +  0][kk + koff + 1];
      b1.x = Ws[oc + 16][kk + koff]; b1.y = Ws[oc + 16][kk + koff + 1];
      b2.x = Ws[oc + 32][kk + koff]; b2.y = Ws[oc + 32][kk + koff + 1];
      b3.x = Ws[oc + 48][kk + koff]; b3.y = Ws[oc + 48][kk + koff + 1];

      acc0 = __builtin_amdgcn_wmma_f32_16x16x4_f32(false, a, false, b0, (short)0, acc0, false, false);
      acc1 = __builtin_amdgcn_wmma_f32_16x16x4_f32(false, a, false, b1, (short)0, acc1, false, false);
      acc2 = __builtin_amdgcn_wmma_f32_16x16x4_f32(false, a, false, b2, (short)0, acc2, false, false);
      acc3 = __builtin_amdgcn_wmma_f32_16x16x4_f32(false, a, false, b3, (short)0, acc3, false, false);
    }
    __syncthreads();
  }

  // epilogue: BN + scattered stores.  C/D layout: reg r -> l = r (+8 for lanes>=16)
  const int lbase = l0 + wv * 16 + ((ln < 16) ? 0 : 8);
  v8f* accs[4] = {&acc0, &acc1, &acc2, &acc3};
#pragma unroll
  for (int tt = 0; tt < 4; ++tt) {
    int o = o0 + tt * 16 + (ln & 15);
    float g  = bn[o];
    float be = bn[O + o];
    float m  = bn[2 * O + o];
    float vv = bn[3 * O + o];
    float s  = g * rsqrtf(vv + BN_EPS);
    float sh = be - m * s;
    v8f a = *accs[tt];
#pragma unroll
    for (int r = 0; r < 8; ++r) {
      float val = a[r] * s + sh;
      size_t idx = ((size_t)b * Lp + (size_t)(lbase + r)) * O + o;
      Y[idx] = val;
      if (Y2) Y2[idx] = val;
    }
  }
}

// ---------------------------------------------------------------------------
// one log-attention step: softmax over groups of `bs` positions at stride `st`
// qk: (BB, Lp, 2*CC) channel order (h, {q,k}, d); v2: (BB, Lp, CC) in-place.
// block = 256 threads = 8 waves; wave w handles head w; 2 d-elems per lane.
// ---------------------------------------------------------------------------
__global__ void __launch_bounds__(256) attn_step(
    const float* __restrict__ qk, float* __restrict__ v2,
    int Lp, int bs, int st)
{
  const int g  = blockIdx.x;       // group id within batch
  const int b  = blockIdx.y;
  const int h  = threadIdx.x >> 5;
  const int ln = threadIdx.x & 31;

  const int l_low  = g % st;
  const int l_high = g / st;
  const size_t lbase = (size_t)l_high * (size_t)st * bs + l_low;

  const float* qkb = qk + (size_t)b * Lp * (2 * CC);
  float*       v2b = v2 + (size_t)b * Lp * CC;
  const int d0 = ln * 2;

  float qx[3], qy[3], kx[3], ky[3], vx[3], vy[3];
  for (int j = 0; j < bs; ++j) {
    size_t l = lbase + (size_t)j * st;
    const float* qptr = qkb + l * (2 * CC) + h * (2 * DD) + d0;
    qx[j] = qptr[0];       qy[j] = qptr[1];
    kx[j] = qptr[DD];      ky[j] = qptr[DD + 1];
    const float* vptr = v2b + l * CC + h * DD + d0;
    vx[j] = vptr[0];       vy[j] = vptr[1];
  }

  float p[3][3];
  for (int jq = 0; jq < bs; ++jq)
    for (int jk = 0; jk < bs; ++jk) {
      float d = qx[jq] * kx[jk] + qy[jq] * ky[jk];
      d += __shfl_xor(d, 16, 32);
      d += __shfl_xor(d,  8, 32);
      d += __shfl_xor(d,  4, 32);
      d += __shfl_xor(d,  2, 32);
      d += __shfl_xor(d,  1, 32);
      p[jq][jk] = d * 0.125f;   // scale = d^-0.5 = 1/8
    }

  for (int jq = 0; jq < bs; ++jq) {
    float mx = p[jq][0];
    for (int jk = 1; jk < bs; ++jk) mx = fmaxf(mx, p[jq][jk]);
    float sum = 0.f;
    for (int jk = 0; jk < bs; ++jk) { p[jq][jk] = expf(p[jq][jk] - mx); sum += p[jq][jk]; }
    float inv = 1.f / sum;
    float ox = 0.f, oy = 0.f;
    for (int jk = 0; jk < bs; ++jk) { ox += p[jq][jk] * vx[jk]; oy += p[jq][jk] * vy[jk]; }
    size_t l = lbase + (size_t)jq * st;
    float* vptr = v2b + l * CC + h * DD + d0;
    vptr[0] = ox * inv;
    vptr[1] = oy * inv;
  }
}

// ---------------------------------------------------------------------------
// xt[b,l,c] += v2[b,l,c] + BN_c(depthwise conv3 of v0 along l)
// ---------------------------------------------------------------------------
__global__ void pe_residual(
    float* __restrict__ xt, const float* __restrict__ v2,
    const float* __restrict__ v0, const float* __restrict__ pw,
    const float* __restrict__ pbn, int Lp, int Wp)
{
  size_t i = (size_t)blockIdx.x * blockDim.x + threadIdx.x;
  const size_t total = (size_t)BB * W0L * CC;
  if (i >= total) return;
  int c = (int)(i % CC);
  int l = (int)((i / CC) % W0L);
  int b = (int)(i / ((size_t)CC * W0L));
  size_t rowb = ((size_t)b * Lp + l) * CC + c;
  float mid   = v0[rowb];
  float left  = (l >= 1)     ? v0[rowb - CC] : 0.f;
  float right = (l + 1 < Wp) ? v0[rowb + CC] : 0.f;
  float conv = left * pw[c * 3 + 0] + mid * pw[c * 3 + 1] + right * pw[c * 3 + 2];
  float g  = pbn[c];
  float be = pbn[CC + c];
  float m  = pbn[2 * CC + c];
  float vv = pbn[3 * CC + c];
  float pe = (conv - m) * (g * rsqrtf(vv + BN_EPS)) + be;
  xt[((size_t)b * W0L + l) * CC + c] += v2[rowb] + pe;
}

// ---------------------------------------------------------------------------
extern "C" void kernel_launch(void* const* d_in, const int* in_sizes, int n_in,
                              void* d_out, int out_size, void* d_ws, size_t ws_size,
                              hipStream_t stream)
{
  (void)in_sizes; (void)n_in; (void)out_size; (void)ws_size;

  const float* x       = (const float*)d_in[0];   // (4,4096,512)
  const float* qk_w    = (const float*)d_in[1];   // (2,1024,512)
  const float* qk_bn   = (const float*)d_in[2];   // (2,4,1024)
  const float* v_w     = (const float*)d_in[3];   // (2,512,512)
  const float* v_bn    = (const float*)d_in[4];   // (2,4,512)
  const float* pe_w    = (const float*)d_in[5];   // (2,512,3)
  const float* pe_bn   = (const float*)d_in[6];   // (2,4,512)
  const float* proj_w  = (const float*)d_in[7];   // (512,512)
  const float* proj_bn = (const float*)d_in[8];   // (4,512)

  float* xt    = (float*)d_ws;                           // (4,4096,512)
  float* xp3   = xt    + (size_t)BB * W0L * CC;          // (4,6656,512)
  float* qkbuf = xp3   + (size_t)BB * LP3 * CC;          // (4,6656,1024)
  float* v0    = qkbuf + (size_t)BB * LP3 * (2 * CC);    // (4,6656,512)
  float* v2    = v0    + (size_t)BB * LP3 * CC;          // (4,6656,512)

  // xt = x  (already (b,l,c) layout)
  {
    int n4 = BB * W0L * CC / 4;
    copy_f4<<<(n4 + 255) / 256, 256, 0, stream>>>((float4*)xt, (const float4*)x, n4);
  }

  // ---------------- base 2 (n=12, Wp=4096) ----------------
  gemm_bn<<<dim3(LP2 / 128, 1024 / 64, BB), 256, 0, stream>>>(
      xt, qk_w, qk_bn, qkbuf, nullptr, LP2, 1024);
  gemm_bn<<<dim3(LP2 / 128, 512 / 64, BB), 256, 0, stream>>>(
      xt, v_w, v_bn, v0, v2, LP2, 512);
  for (int i = 0; i < N2; ++i) {
    int st = 1 << (N2 - 1 - i);
    attn_step<<<dim3(WP2 / 2, BB), 256, 0, stream>>>(qkbuf, v2, LP2, 2, st);
  }
  pe_residual<<<(int)(((size_t)BB * W0L * CC + 255) / 256), 256, 0, stream>>>(
      xt, v2, v0, pe_w, pe_bn, LP2, WP2);

  // ---------------- base 3 (n=8, Wp=6561) ----------------
  {
    size_t n4 = (size_t)BB * LP3 * CC / 4;
    pad_rows<<<(int)((n4 + 255) / 256), 256, 0, stream>>>((float4*)xp3, (const float4*)xt);
  }
  gemm_bn<<<dim3(LP3 / 128, 1024 / 64, BB), 256, 0, stream>>>(
      xp3, qk_w + 1024 * 512, qk_bn + 4 * 1024, qkbuf, nullptr, LP3, 1024);
  gemm_bn<<<dim3(LP3 / 128, 512 / 64, BB), 256, 0, stream>>>(
      xp3, v_w + 512 * 512, v_bn + 4 * 512, v0, v2, LP3, 512);
  {
    int st = 2187;  // 3^7
    for (int i = 0; i < N3; ++i) {
      attn_step<<<dim3(WP3 / 3, BB), 256, 0, stream>>>(qkbuf, v2, LP3, 3, st);
      st /= 3;
    }
  }
  pe_residual<<<(int)(((size_t)BB * W0L * CC + 255) / 256), 256, 0, stream>>>(
      xt, v2, v0, pe_w + 512 * 3, pe_bn + 4 * 512, LP3, WP3);

  // ---------------- final projection -> d_out (b,l,o) = (4,4096,512) ----------------
  gemm_bn<<<dim3(W0L / 128, 512 / 64, BB), 256, 0, stream>>>(
      xt, proj_w, proj_bn, (float*)d_out, nullptr, W0L, 512);
}